// Lifter_62466004353136
// MI455X (gfx1250) — compile-verified
//
#include <hip/hip_runtime.h>
#include <hip/hip_bf16.h>

// Problem constants from the reference
#define B_   2
#define N_   8
#define C_   64
#define H_   192
#define W_   256
#define V_   200000
#define R_   (N_*H_*W_)      // 393216 rays per batch
#define OUT_ 128
#define K_   (C_+1)          // 65 (features + confidence)
#define KPAD 68              // 17 chunks of 4 for WMMA f32 16x16x4; 68 % 64 == 4 -> LDS conflict-free
#define NKC  17
#define MTILES ((B_*V_)/16)  // 25000
#define GEMM_BLOCKS 2500     // persistent blocks; each handles 10 M tiles

typedef __attribute__((ext_vector_type(2))) float v2f;
typedef __attribute__((ext_vector_type(8))) float v8f;

// ---------------------------------------------------------------------------
// Ray directions: d = normalize(R[:, :3] . [uu, vv, 1])
// (pixel_xyz - origin cancels the translation column of the pose)
// ---------------------------------------------------------------------------
__global__ void rays_kernel(const float* __restrict__ pose,
                            const float* __restrict__ intr,
                            float* __restrict__ dirs) {
    int t = blockIdx.x * blockDim.x + threadIdx.x;   // over B*N*H*W
    if (t >= B_ * N_ * H_ * W_) return;
    int w  = t % W_;
    int h  = (t / W_) % H_;
    int bn = t / (W_ * H_);

    const float* I = intr + bn * 6;
    float uu = ((float)w - I[2]) / I[0];
    float vv = ((float)h - I[3]) / I[1];

    const float* P = pose + bn * 16;                 // 4x4 row-major
    float dx = P[0]*uu + P[1]*vv + P[2];
    float dy = P[4]*uu + P[5]*vv + P[6];
    float dz = P[8]*uu + P[9]*vv + P[10];
    float inv = rsqrtf(dx*dx + dy*dy + dz*dz);

    long o = (long)t * 3;
    dirs[o + 0] = dx * inv;
    dirs[o + 1] = dy * inv;
    dirs[o + 2] = dz * inv;
}

// ---------------------------------------------------------------------------
// Per-voxel hit counts (one thread per ray)
// ---------------------------------------------------------------------------
__global__ void count_kernel(const int* __restrict__ vids,
                             const int* __restrict__ mask,
                             float* __restrict__ counts) {
    int t = blockIdx.x * blockDim.x + threadIdx.x;   // over B*R
    if (t >= B_ * R_) return;
    if (mask[t]) {
        int b = t / R_;
        atomicAdd(&counts[(long)b * V_ + vids[t]], 1.0f);
    }
}

// ---------------------------------------------------------------------------
// counts -> 1 / max(counts, 1) in place, so the GEMM only multiplies.
// ---------------------------------------------------------------------------
__global__ void inv_kernel(float* __restrict__ counts) {
    int t = blockIdx.x * blockDim.x + threadIdx.x;   // over B*V
    if (t >= B_ * V_) return;
    counts[t] = 1.0f / fmaxf(counts[t], 1.0f);
}

// ---------------------------------------------------------------------------
// Scatter-add features into voxel sums. One thread per feature element in the
// native (b, n, c, h*w) layout -> fully coalesced feature reads; voxel_ids /
// hit_mask (6 MB) live in L2.
// ---------------------------------------------------------------------------
__global__ void scatter_kernel(const float* __restrict__ feat,
                               const int* __restrict__ vids,
                               const int* __restrict__ mask,
                               float* __restrict__ sums) {
    long t = (long)blockIdx.x * blockDim.x + threadIdx.x;  // over B*N*C*H*W
    if (t >= (long)B_ * N_ * C_ * H_ * W_) return;
    int  hw = (int)(t % (H_ * W_));
    long q  = t / (H_ * W_);
    int  c  = (int)(q % C_);
    long bn = q / C_;
    int  n  = (int)(bn % N_);
    int  b  = (int)(bn / N_);

    int br = b * R_ + n * (H_ * W_) + hw;
    if (mask[br]) {
        int vid = vids[br];
        atomicAdd(&sums[((long)(b * V_ + vid)) * C_ + c], feat[t]);
    }
}

// ---------------------------------------------------------------------------
// Fused finalize + GEMM with V_WMMA_F32_16X16X4_F32.
// M = B*V rows, N = 128 cols, K = 65 padded to 68 (17 WMMA k-steps).
// Persistent: 2500 blocks x 256 threads (8 wave32 waves). Per M tile, the
// block stages a pre-scaled 16xKPAD A tile in LDS (coalesced, done once, not
// 8x per wave); wave w computes the 16x16 tile for columns [16w, 16w+16).
// w_fc is staged in LDS once per block.
// ---------------------------------------------------------------------------
__global__ __launch_bounds__(256)
void gemm_kernel(const float* __restrict__ sums,
                 const float* __restrict__ invcnt,   // 1/max(cnt,1), precomputed
                 const float* __restrict__ conf,
                 const float* __restrict__ w_fc,
                 const float* __restrict__ b_fc,
                 float* __restrict__ out) {
    __shared__ float lds_w[KPAD * OUT_];   // 34816 B, zero-padded K
    __shared__ float lds_a[16 * KPAD];     //  4352 B, pre-scaled A tile

    for (int i = threadIdx.x; i < KPAD * OUT_; i += 256) {
        int k = i / OUT_;
        lds_w[i] = (k < K_) ? w_fc[i] : 0.0f;        // same flat index when k < K_
    }

    int lane = threadIdx.x & 31;
    int wave = threadIdx.x >> 5;                     // N tile id, 0..7
    int half = lane >> 4;                            // selects K pair (A) / K rows (B)
    int rlow = lane & 15;
    int nc   = wave * 16 + rlow;                     // output column for B / D
    float bias = b_fc[nc];

    for (int mt = blockIdx.x; mt < MTILES; mt += GEMM_BLOCKS) {
        __syncthreads();   // lds_w ready (1st iter) / lds_a no longer read (later iters)

        // Stage A tile: 16 rows x 64 ch, coalesced, scaled by 1/max(cnt,1)
        long gbase = (long)mt * 16;
        for (int i = threadIdx.x; i < 16 * C_; i += 256) {
            int r = i >> 6;                          // i / 64
            int k = i & 63;
            lds_a[r * KPAD + k] = sums[(gbase + r) * C_ + k] * invcnt[gbase + r];
        }
        // Confidence column + zero K padding (cols 64..67)
        for (int i = threadIdx.x; i < 16 * (KPAD - C_); i += 256) {
            int r = i / (KPAD - C_);
            int k = C_ + i % (KPAD - C_);
            lds_a[r * KPAD + k] = (k == C_) ? conf[gbase + r] : 0.0f;
        }
        __syncthreads();

        v8f acc = {};
        for (int kc = 0; kc < NKC; ++kc) {
            // A 16x4 f32 layout: lanes 0-15 hold K={0,1}, lanes 16-31 hold K={2,3}
            int k0 = kc * 4 + half * 2;
            v2f a, bf;
            a.x = lds_a[rlow * KPAD + k0];           // row stride 68 -> conflict-free
            a.y = lds_a[rlow * KPAD + k0 + 1];
            // B 4x16 f32: row K striped across 16 lanes per half-wave
            bf.x = lds_w[k0 * OUT_ + nc];
            bf.y = lds_w[(k0 + 1) * OUT_ + nc];
            acc = __builtin_amdgcn_wmma_f32_16x16x4_f32(
                /*neg_a=*/false, a, /*neg_b=*/false, bf,
                /*c_mod=*/(short)0, acc, /*reuse_a=*/false, /*reuse_b=*/false);
        }

        // D 16x16 f32 layout: VGPR j -> M = j + 8*half, N = lane&15 (in N tile)
#pragma unroll
        for (int j = 0; j < 8; ++j) {
            long row = gbase + j + 8 * half;
            out[row * OUT_ + nc] = acc[j] + bias;
        }
    }
}

// ---------------------------------------------------------------------------
extern "C" void kernel_launch(void* const* d_in, const int* in_sizes, int n_in,
                              void* d_out, int out_size, void* d_ws, size_t ws_size,
                              hipStream_t stream) {
    const float* pose = (const float*)d_in[0];   // (B,N,4,4)
    const float* intr = (const float*)d_in[1];   // (B,N,6)
    const float* feat = (const float*)d_in[2];   // (B,N,C,H,W)
    // d_in[3] = depths (unused by reference outputs)
    const float* conf = (const float*)d_in[4];   // (B,V,1)
    const int*   vids = (const int*)d_in[5];     // (B,R) int32
    const int*   mask = (const int*)d_in[6];     // (B,R) bool -> int
    const float* w_fc = (const float*)d_in[7];   // (C+1, OUT)
    const float* b_fc = (const float*)d_in[8];   // (OUT,)

    float* out  = (float*)d_out;                         // (B,V,OUT)
    float* dirs = out + (size_t)B_ * V_ * OUT_;          // (B,N,H,W,3)

    float* sums   = (float*)d_ws;                        // (B,V,C)
    float* counts = sums + (size_t)B_ * V_ * C_;         // (B,V) -> inv after inv_kernel
    size_t zbytes = ((size_t)B_ * V_ * C_ + (size_t)B_ * V_) * sizeof(float);
    hipMemsetAsync(d_ws, 0, zbytes, stream);

    rays_kernel<<<(B_ * N_ * H_ * W_ + 255) / 256, 256, 0, stream>>>(pose, intr, dirs);
    count_kernel<<<(B_ * R_ + 255) / 256, 256, 0, stream>>>(vids, mask, counts);
    inv_kernel<<<(B_ * V_ + 255) / 256, 256, 0, stream>>>(counts);
    long nfeat = (long)B_ * N_ * C_ * H_ * W_;
    scatter_kernel<<<(int)((nfeat + 255) / 256), 256, 0, stream>>>(feat, vids, mask, sums);
    gemm_kernel<<<GEMM_BLOCKS, 256, 0, stream>>>(sums, counts, conf, w_fc, b_fc, out);
}